// DirectVoxGO_29308856828732
// MI455X (gfx1250) — compile-verified
//
#include <hip/hip_runtime.h>

// DirectVoxGO Raw2Alpha + Alphas2Weights, wave32 segmented-scan formulation.
//
// ALPHA_INIT = 1e-4  ->  SHIFT = log(1/(1-1e-4) - 1) = -9.2102403669758...
// INTERVAL   = 0.5
//
// sp    = softplus(density + SHIFT)
// alpha = 1 - exp(-0.5*sp) = 1 - rsqrt(1 + exp(z))       (z = density+SHIFT)
// T_i   = exp(-0.5 * exclusive_segment_cumsum(sp))
// w_i   = alpha_i * T_i
// alphainv_last[r] = exp(-0.5 * segment_sum(sp))
//
// ray_id is sorted -> segment r = [lower_bound(r), lower_bound(r+1)).
// One wave per ray. Streams (density/weights) are non-temporal B128 so the
// ray_id binary-search tree stays hot in the 192MB L2; 128 elements per wave
// iteration; ds_bpermute wave scan of lane partial sums; readlane carry.

#define DVGO_SHIFT (-9.2102403669758f)

typedef float v4f __attribute__((ext_vector_type(4)));

__device__ __forceinline__ float bcast_f32(float v, int srcLane) {
  return __int_as_float(__builtin_amdgcn_readlane(__float_as_int(v), srcLane));
}

__device__ __forceinline__ int bcast_i32(int v, int srcLane) {
  return __builtin_amdgcn_readlane(v, srcLane);
}

// Inclusive wave32 add-scan via ds_bpermute (LDS crossbar, no LDS storage).
__device__ __forceinline__ float wave_incl_scan_add(float v, int lane) {
#pragma unroll
  for (int d = 1; d < 32; d <<= 1) {
    float n = __int_as_float(
        __builtin_amdgcn_ds_bpermute(((lane - d) & 31) << 2, __float_as_int(v)));
    v += (lane >= d) ? n : 0.0f;
  }
  return v;
}

// sp = softplus(z) (stable), alpha = 1 - (1+exp(z))^-1/2  == 1 - exp(-0.5*sp)
__device__ __forceinline__ void sp_alpha(float z, float& sp, float& alpha) {
  float t = __expf(z);                       // v_exp_f32
  sp = (z > 15.0f) ? z : __logf(1.0f + t);   // v_log_f32 (log1p ok: t>=~3e6 skipped)
  alpha = 1.0f - __frsqrt_rn(1.0f + t);      // v_rsq_f32
}

// First index i in [0,m) with a[i] >= key (a sorted ascending). Regular
// (cacheable) loads on purpose: all waves share the upper tree levels in L2.
__device__ __forceinline__ int lower_bound_i32(const int* __restrict__ a, int m,
                                               int key) {
  int lo = 0, hi = m;
  while (lo < hi) {
    int mid = (lo + hi) >> 1;
    if (a[mid] < key) lo = mid + 1; else hi = mid;
  }
  return lo;
}

__global__ void __launch_bounds__(256) dvgo_weights_kernel(
    const float* __restrict__ density, const int* __restrict__ ray_id,
    float* __restrict__ weights, float* __restrict__ alphainv_last,
    int M, int N) {
  const int lane = (int)(threadIdx.x & 31u);
  const int waveInBlock = (int)(threadIdx.x >> 5);
  const int r = (int)blockIdx.x * (int)(blockDim.x >> 5) + waveInBlock;
  if (r >= N) return;

  // Lanes 0/1 search for r and r+1 in parallel; broadcast both bounds.
  const int pos = lower_bound_i32(ray_id, M, r + (lane & 1));
  const int lo = bcast_i32(pos, 0);
  const int hi = bcast_i32(pos, 1);

  float carry = 0.0f;  // running sum of sp over this ray's processed prefix

  // ---- prologue: <=3 scalar elements to reach a 16B boundary ----
  int lo_al = (lo + 3) & ~3;
  if (lo_al > hi) lo_al = hi;
  if (lo < lo_al) {
    const int i = lo + lane;
    const bool act = i < lo_al;
    float z = act ? (__builtin_nontemporal_load(density + i) + DVGO_SHIFT) : 0.0f;
    float sp, alpha;
    sp_alpha(z, sp, alpha);
    sp = act ? sp : 0.0f;
    float incl = wave_incl_scan_add(sp, lane);
    float excl = carry + (incl - sp);
    if (act)
      __builtin_nontemporal_store(alpha * __expf(-0.5f * excl), weights + i);
    carry += bcast_f32(incl, 31);
  }

  // ---- main: 128 elements / iteration, float4 per lane (B128 NT stream) ----
  int base = lo_al;
  for (; base + 128 <= hi; base += 128) {
    if (base + 256 <= hi)  // global_prefetch_b8 one vector-iter ahead
      __builtin_prefetch(density + base + 128 + 4 * lane, 0, 1);

    const v4f d4 =
        __builtin_nontemporal_load((const v4f*)(density + base) + lane);

    float sp0, sp1, sp2, sp3, a0, a1, a2, a3;
    sp_alpha(d4.x + DVGO_SHIFT, sp0, a0);
    sp_alpha(d4.y + DVGO_SHIFT, sp1, a1);
    sp_alpha(d4.z + DVGO_SHIFT, sp2, a2);
    sp_alpha(d4.w + DVGO_SHIFT, sp3, a3);

    const float laneSum = (sp0 + sp1) + (sp2 + sp3);
    const float incl = wave_incl_scan_add(laneSum, lane);
    const float e0 = carry + (incl - laneSum);
    const float e1 = e0 + sp0;
    const float e2 = e1 + sp1;
    const float e3 = e2 + sp2;

    v4f w;
    w.x = a0 * __expf(-0.5f * e0);
    w.y = a1 * __expf(-0.5f * e1);
    w.z = a2 * __expf(-0.5f * e2);
    w.w = a3 * __expf(-0.5f * e3);
    __builtin_nontemporal_store(w, (v4f*)(weights + base) + lane);

    carry += bcast_f32(incl, 31);
  }

  // ---- tail: one masked component-wise iteration (< 128 elements) ----
  if (base < hi) {
    const int i0 = base + 4 * lane;
    float sp[4], al[4];
#pragma unroll
    for (int j = 0; j < 4; ++j) {
      const int idx = i0 + j;
      const bool act = idx < hi;
      float z = act ? (__builtin_nontemporal_load(density + idx) + DVGO_SHIFT)
                    : 0.0f;
      sp_alpha(z, sp[j], al[j]);
      sp[j] = act ? sp[j] : 0.0f;
    }
    const float laneSum = (sp[0] + sp[1]) + (sp[2] + sp[3]);
    const float incl = wave_incl_scan_add(laneSum, lane);
    float e = carry + (incl - laneSum);
#pragma unroll
    for (int j = 0; j < 4; ++j) {
      const int idx = i0 + j;
      if (idx < hi)
        __builtin_nontemporal_store(al[j] * __expf(-0.5f * e), weights + idx);
      e += sp[j];
    }
    carry += bcast_f32(incl, 31);
  }

  if (lane == 0) alphainv_last[r] = __expf(-0.5f * carry);
}

extern "C" void kernel_launch(void* const* d_in, const int* in_sizes, int n_in,
                              void* d_out, int out_size, void* d_ws,
                              size_t ws_size, hipStream_t stream) {
  const float* density = (const float*)d_in[0];
  const int*   ray_id  = (const int*)d_in[1];
  const int M = in_sizes[0];
  const int N = out_size - M;  // outputs: weights[M] ++ alphainv_last[N]

  float* weights  = (float*)d_out;
  float* alphainv = (float*)d_out + M;

  const int threads = 256;  // 8 waves per block (wave32)
  const int wavesPerBlock = threads / 32;
  const int blocks = (N + wavesPerBlock - 1) / wavesPerBlock;

  dvgo_weights_kernel<<<blocks, threads, 0, stream>>>(
      density, ray_id, weights, alphainv, M, N);
}